// SSIM_6322191860385
// MI455X (gfx1250) — compile-verified
//
#include <hip/hip_runtime.h>

// ---------------------------------------------------------------------------
// SSIM on MI455X (gfx1250): separable 11x11 Gaussian convs as banded GEMMs on
// V_WMMA_F32_16X16X4_F32 (fp32 WMMA — sigma = E[x^2]-mu^2 cancels, so low
// precision shapes are out). One wave32 per 16x16 output tile.
// LDS weight table (v_exp_f32), 16B-aligned B128 staging with
// GLOBAL_LOAD_ASYNC_TO_LDS_B128 for interior tiles, overlapped row-tiles to
// avoid all masking in the WMMA passes, wave barriers for wave-private LDS,
// v_rcp for the SSIM divides, fp64 atomic reduction.
// ---------------------------------------------------------------------------

typedef __attribute__((ext_vector_type(2))) float v2f;
typedef __attribute__((ext_vector_type(4))) float v4f;
typedef __attribute__((ext_vector_type(8))) float v8f;
typedef int ai4 __attribute__((vector_size(4 * sizeof(int))));

#define GLOBAL_AS __attribute__((address_space(1)))
#define LDS_AS __attribute__((address_space(3)))

#define WAVES 4
#define IMG 512
#define NTILES_TOTAL (48 * 32 * 32)   // 49152 tiles of 16x16
#define NELEM_D 12582912.0            // 16*3*512*512

#if __has_builtin(__builtin_amdgcn_global_load_async_to_lds_b128)
#define HAVE_ASYNC_LDS 1
#else
#define HAVE_ASYNC_LDS 0
#endif

#if HAVE_ASYNC_LDS
__device__ __forceinline__ void wait_async_lds() {
#if __has_builtin(__builtin_amdgcn_s_wait_asynccnt)
  __builtin_amdgcn_s_wait_asynccnt(0);
#else
  asm volatile("s_wait_asynccnt 0x0" ::: "memory");
#endif
}
__device__ __forceinline__ void async_b128(const float* g, float* l) {
  __builtin_amdgcn_global_load_async_to_lds_b128((GLOBAL_AS ai4*)g,
                                                 (LDS_AS ai4*)l, 0, 0);
}
#endif

__device__ __forceinline__ v8f wmma4(v2f a, v2f b, v8f c) {
  // D(16x16,f32) = A(16x4,f32) * B(4x16,f32) + C
  return __builtin_amdgcn_wmma_f32_16x16x4_f32(false, a, false, b,
                                               (short)0, c, false, false);
}

__global__ void ssim_zero_accum(double* accum) { accum[0] = 0.0; }

__global__ __launch_bounds__(32 * WAVES) void ssim_map_kernel(
    const float* __restrict__ img1, const float* __restrict__ img2,
    double* __restrict__ accum) {
  // Wave-private LDS slices + one block-shared weight table.
  __shared__ float s_in1[WAVES][26][32];     // img1 halo tile (x0-8..x0+23)
  __shared__ float s_in2[WAVES][26][32];     // img2 halo tile
  __shared__ float s_mid[WAVES][5][28][16];  // 5 horizontal conv results
  __shared__ float s_wtab[52];               // w[d] at index d+20, d in [0,10]
  __shared__ float s_red[WAVES];

  const int lane = threadIdx.x & 31;
  const int wv = threadIdx.x >> 5;
  const int tile = blockIdx.x * WAVES + wv;  // < 49152
  const int p = tile >> 10;                  // plane (batch*channel)
  const int rem = tile & 1023;
  const int ty = rem >> 5;
  const int tx = rem & 31;
  const int y0 = ty * 16, x0 = tx * 16;
  const size_t base = (size_t)p * (IMG * IMG);

  // ---- Gaussian weight table, built once per block via v_exp_f32 ----
  // w[d] = (1/sum) * exp(-(d-5)^2 / (2*1.5^2)), window 11, sigma 1.5.
  if (threadIdx.x < 52) {
    const int d = (int)threadIdx.x - 20;
    const float u = (float)(d - 5);
    const float val = 0.26601145f * exp2f(-0.32059889f * u * u);
    s_wtab[threadIdx.x] = ((unsigned)d <= 10u) ? val : 0.f;
  }

  // ---- Zero mid rows 26,27 once (vertical K tail must read zeros) ----
  {
    const v4f z = {};
#pragma unroll
    for (int zi = 0; zi < 2; ++zi) {
      const int idx = zi * 32 + lane;        // 40 float4 = 5 sig * 2 rows * 16
      if (idx < 40) {
        const int s = idx >> 3, f = (idx & 7) * 4;
        *(v4f*)&s_mid[wv][s][26][f] = z;
      }
    }
  }

  // ---- Stage 26x32 halo tiles (16B aligned: x0-8 is a float4 boundary) ----
  // Per-lane base: row r0 = lane>>3, float4 col c4 = lane&7; advance 4 rows
  // per iteration (6 full iterations = rows 0..23, half-wave tail rows 24,25).
  const int r0 = lane >> 3;
  const int c4 = lane & 7;
  const int gx = x0 - 8 + 4 * c4;
  const float* p1 = img1 + base;
  const float* p2 = img2 + base;
  const bool interior = (tx >= 1) && (tx <= 30) && (ty >= 1) && (ty <= 30);

#if HAVE_ASYNC_LDS
  if (interior) {
    const float* g1 = p1 + (size_t)(y0 - 5 + r0) * IMG + gx;
    const float* g2 = p2 + (size_t)(y0 - 5 + r0) * IMG + gx;
    float* l1 = &s_in1[wv][r0][4 * c4];
    float* l2 = &s_in2[wv][r0][4 * c4];
#pragma unroll
    for (int it = 0; it < 6; ++it) {
      async_b128(g1 + (size_t)it * 4 * IMG, l1 + it * 4 * 32);
      async_b128(g2 + (size_t)it * 4 * IMG, l2 + it * 4 * 32);
    }
    if (lane < 16) {  // rows 24,25
      async_b128(g1 + (size_t)24 * IMG, l1 + 24 * 32);
      async_b128(g2 + (size_t)24 * IMG, l2 + 24 * 32);
    }
    wait_async_lds();
  } else
#endif
  {
    // Border (or no-async) path: masked float4 loads, zero padding.
    const bool okx = (unsigned)gx < (unsigned)IMG;  // float4 all-in/all-out
#pragma unroll
    for (int it = 0; it < 7; ++it) {
      const int r = 4 * it + r0;             // tail: rows 24,25 on lanes<16
      if (it < 6 || lane < 16) {
        const int gy = y0 - 5 + r;
        const bool ok = okx && ((unsigned)gy < (unsigned)IMG);
        v4f a = {}, b = {};
        if (ok) {
          const size_t o = (size_t)gy * IMG + gx;
          a = *(const v4f*)(p1 + o);
          b = *(const v4f*)(p2 + o);
        }
        *(v4f*)&s_in1[wv][r][4 * c4] = a;
        *(v4f*)&s_in2[wv][r][4 * c4] = b;
      }
    }
  }
  __syncthreads();  // weight table (cross-wave) + own staging visible

  const int mloc = lane & 15;
  const int hi = lane >> 4;

  // Per-lane banded-weight chunks from the LDS table. Under the wave32
  // 16x16x4 f32 operand layout (vreg0: K=4q+2*hi, vreg1: K+1; M/N=lane&15):
  //   horizontal B value = w[k - n - 3], vertical A value = w[k - m].
  const int aV = 2 * hi - mloc + 20;
  const int aH = aV - 3;
  v2f bh[8], bv[7];
#pragma unroll
  for (int q = 0; q < 8; ++q) {
    bh[q].x = s_wtab[aH + 4 * q];
    bh[q].y = s_wtab[aH + 4 * q + 1];
  }
#pragma unroll
  for (int q = 0; q < 7; ++q) {
    bv[q].x = s_wtab[aV + 4 * q];
    bv[q].y = s_wtab[aV + 4 * q + 1];
  }

  // ---- Horizontal pass: mid_s(26x16) = prod_s(rows x 32) * Wband(32x16) ----
  // Row-tile 0 covers mid rows 0..15, row-tile 1 covers rows 10..25 (the
  // 10..15 overlap recomputes identical values -> idempotent stores, and no
  // masking anywhere). Signals: x1, x2, x1^2, x2^2, x1*x2, built on the fly.
#pragma unroll
  for (int t = 0; t < 2; ++t) {
    const int rb = t ? 10 : 0;
    v8f h0 = {}, h1 = {}, h2 = {}, h3 = {}, h4 = {};
    const int row = rb + mloc;  // input/mid row, always < 26
#pragma unroll
    for (int q = 0; q < 8; ++q) {
      const int k0 = 4 * q + 2 * hi;  // even -> 8B-aligned v2f LDS load
      const v2f u1 = *(const v2f*)&s_in1[wv][row][k0];
      const v2f u2 = *(const v2f*)&s_in2[wv][row][k0];
      v2f a3; a3.x = u1.x * u1.x; a3.y = u1.y * u1.y;
      v2f a4; a4.x = u2.x * u2.x; a4.y = u2.y * u2.y;
      v2f a5; a5.x = u1.x * u2.x; a5.y = u1.y * u2.y;
      h0 = wmma4(u1, bh[q], h0);
      h1 = wmma4(u2, bh[q], h1);
      h2 = wmma4(a3, bh[q], h2);
      h3 = wmma4(a4, bh[q], h3);
      h4 = wmma4(a5, bh[q], h4);
    }
    // D layout: vreg g -> (M = g + 8*hi, N = lane&15); rows rb..rb+15.
    const int rbase = rb + 8 * hi;
#pragma unroll
    for (int g = 0; g < 8; ++g) {
      const int ro = rbase + g;
      s_mid[wv][0][ro][mloc] = h0[g];
      s_mid[wv][1][ro][mloc] = h1[g];
      s_mid[wv][2][ro][mloc] = h2[g];
      s_mid[wv][3][ro][mloc] = h3[g];
      s_mid[wv][4][ro][mloc] = h4[g];
    }
  }
  // Wave-private LDS; DS ops from one wave are in-order (DScnt) — only a
  // compiler scheduling fence is needed, not a block barrier.
  __builtin_amdgcn_wave_barrier();

  // ---- Vertical pass: out_s(16x16) = Wband(16x28) * mid_s(28x16) ----
  v8f o0 = {}, o1 = {}, o2 = {}, o3 = {}, o4 = {};
#pragma unroll
  for (int q = 0; q < 7; ++q) {
    const int k0 = 4 * q + 2 * hi;
    v2f b0, b1, b2, b3, b4;
    b0.x = s_mid[wv][0][k0][mloc]; b0.y = s_mid[wv][0][k0 + 1][mloc];
    b1.x = s_mid[wv][1][k0][mloc]; b1.y = s_mid[wv][1][k0 + 1][mloc];
    b2.x = s_mid[wv][2][k0][mloc]; b2.y = s_mid[wv][2][k0 + 1][mloc];
    b3.x = s_mid[wv][3][k0][mloc]; b3.y = s_mid[wv][3][k0 + 1][mloc];
    b4.x = s_mid[wv][4][k0][mloc]; b4.y = s_mid[wv][4][k0 + 1][mloc];
    o0 = wmma4(bv[q], b0, o0);
    o1 = wmma4(bv[q], b1, o1);
    o2 = wmma4(bv[q], b2, o2);
    o3 = wmma4(bv[q], b3, o3);
    o4 = wmma4(bv[q], b4, o4);
  }

  // ---- SSIM map (layout-agnostic: all 5 accumulators share D layout) ----
  const float C1 = 0.0001f, C2 = 0.0009f;
  float lsum = 0.f;
#pragma unroll
  for (int g = 0; g < 8; ++g) {
    const float mu1 = o0[g], mu2 = o1[g];
    const float e11 = o2[g], e22 = o3[g], e12 = o4[g];
    const float mu1s = mu1 * mu1, mu2s = mu2 * mu2, mu12 = mu1 * mu2;
    const float sg1 = e11 - mu1s, sg2 = e22 - mu2s, sg12 = e12 - mu12;
    const float num = (2.f * mu12 + C1) * (2.f * sg12 + C2);
    const float den = (mu1s + mu2s + C1) * (sg1 + sg2 + C2);
    lsum = fmaf(num, __builtin_amdgcn_rcpf(den), lsum);
  }

  // ---- Reduce: wave shuffle -> LDS -> one fp64 atomic per block ----
#pragma unroll
  for (int off = 16; off > 0; off >>= 1) lsum += __shfl_xor(lsum, off, 32);
  if (lane == 0) s_red[wv] = lsum;
  __syncthreads();
  if (threadIdx.x == 0) {
    float bs = 0.f;
#pragma unroll
    for (int i = 0; i < WAVES; ++i) bs += s_red[i];
    atomicAdd(accum, (double)bs);
  }
}

__global__ void ssim_finalize(const double* __restrict__ accum,
                              float* __restrict__ out) {
  out[0] = 1.0f - (float)(accum[0] / NELEM_D);
}

extern "C" void kernel_launch(void* const* d_in, const int* in_sizes, int n_in,
                              void* d_out, int out_size, void* d_ws,
                              size_t ws_size, hipStream_t stream) {
  const float* img1 = (const float*)d_in[0];
  const float* img2 = (const float*)d_in[1];
  float* out = (float*)d_out;
  double* accum = (double*)d_ws;  // d_ws is poisoned -> zero it on-stream

  ssim_zero_accum<<<1, 1, 0, stream>>>(accum);
  ssim_map_kernel<<<NTILES_TOTAL / WAVES, 32 * WAVES, 0, stream>>>(img1, img2,
                                                                   accum);
  ssim_finalize<<<1, 1, 0, stream>>>(accum, out);
}